// HoughSRLoss_57277683859543
// MI455X (gfx1250) — compile-verified
//
#include <hip/hip_runtime.h>
#include <hip/hip_bf16.h>
#include <math.h>

// ---------------- problem constants ----------------
#define ALPHA_F      0.5f
#define NUM_THETA    180
#define RHO_BINS     512
#define LINE_THRESH  50u
#define EPS_F        1e-7f
#define H_IMG        512
#define W_IMG        512
#define N_MAPS       8                       // 4 pred maps + 4 target maps
#define WORDS_PER_ROW (W_IMG / 32)           // 16
#define WORDS_PER_IMG (H_IMG * WORDS_PER_ROW)// 8192
#define MASK_WORDS   (N_MAPS * WORDS_PER_IMG)// 65536
#define HBINS        (RHO_BINS * NUM_THETA)  // 92160
#define GHIST_WORDS  (N_MAPS * HBINS)        // 737280
#define TCHUNK       16
#define NCHUNK       ((NUM_THETA + TCHUNK - 1) / TCHUNK) // 12
#define NBANDS       8
#define ROWS_PER_BAND (H_IMG / NBANDS)       // 64
#define BAND_WORDS   (ROWS_PER_BAND * WORDS_PER_ROW)     // 1024 (4 KB)
#define DIAG_F       724.0773439350246f      // sqrt(512^2+512^2)

typedef float    v2f  __attribute__((ext_vector_type(2)));
typedef float    v8f  __attribute__((ext_vector_type(8)));
typedef unsigned u32x4 __attribute__((ext_vector_type(4)));
typedef int      i32x4 __attribute__((ext_vector_type(4)));
typedef int      i32x8 __attribute__((ext_vector_type(8)));

// ---------------- TDM: 1-D DMA of nwords u32 from global into LDS --------
// D# per CDNA5 ISA ch.8: group0 = {count, lds_addr, global_addr, type=2},
// group1 = {data_size=4B, tensor_dim0 = tile_dim0 = nwords, stride0 = nwords}.
// Groups 2/3 unused (<=2-D tensor). This toolchain's 6-arg builtin form.
__device__ __forceinline__ void tdm_load_1d(unsigned lds_off, const void* gptr,
                                            unsigned nwords) {
    unsigned long long ga = (unsigned long long)(uintptr_t)gptr;
    u32x4 g0;
    g0[0] = 1u;                                           // count=1, user descriptor
    g0[1] = lds_off;                                      // lds_addr (bytes)
    g0[2] = (unsigned)(ga & 0xFFFFFFFFull);               // global_addr[31:0]
    g0[3] = (unsigned)((ga >> 32) & 0x1FFFFFFull)         // global_addr[56:32]
          | (2u << 30);                                   // type = 2 ("image")
    i32x8 g1;
    g1[0] = (int)(2u << 16);                              // data_size=2 -> 4 bytes
    g1[1] = (int)((nwords & 0xFFFFu) << 16);              // tensor_dim0[15:0]
    g1[2] = (int)((nwords >> 16) & 0xFFFFu) | (1 << 16);  // dim0 hi | tensor_dim1=1
    g1[3] = (int)((nwords & 0xFFFFu) << 16);              // tile_dim0 = nwords
    g1[4] = 0;                                            // tile_dim1/2 unused
    g1[5] = (int)nwords;                                  // tensor_dim0_stride lo
    g1[6] = 0;
    g1[7] = 0;
    i32x4 z4 = {0, 0, 0, 0};
    i32x8 z8 = {0, 0, 0, 0, 0, 0, 0, 0};
    __builtin_amdgcn_tensor_load_to_lds(g0, g1, z4, z4, z8, 0);
}

// ---------------- kernel 0: zero histogram + scalar area ----------------
__global__ __launch_bounds__(256) void zero_kernel(unsigned* __restrict__ p, int n) {
    int i = blockIdx.x * 256 + threadIdx.x;
    if (i < n) p[i] = 0u;
}

// ---------------- kernel 1: sigmoid, bitmask pack, image-dice partials ----
__global__ __launch_bounds__(256) void prep_kernel(const float* __restrict__ logits,
                                                   const float* __restrict__ targets,
                                                   unsigned* __restrict__ masks,
                                                   float* __restrict__ scal) {
    const int tid = blockIdx.x * 256 + threadIdx.x;      // 0..32767
    const float4* l4 = (const float4*)logits + (size_t)tid * 8;
    const float4* t4 = (const float4*)targets + (size_t)tid * 8;
    unsigned pb = 0u, tb = 0u;
    float sp = 0.f, st = 0.f, si = 0.f;
#pragma unroll
    for (int k = 0; k < 8; ++k) {
        float4 l = l4[k];
        float4 t = t4[k];
        float lv[4] = {l.x, l.y, l.z, l.w};
        float tv[4] = {t.x, t.y, t.z, t.w};
#pragma unroll
        for (int j = 0; j < 4; ++j) {
            float p = 1.0f / (1.0f + __expf(-lv[j]));    // sigmoid
            sp += p; st += tv[j]; si += p * tv[j];
            int bit = k * 4 + j;
            if (lv[j] > 0.0f)  pb |= (1u << bit);        // sigmoid(l)>0.5 <=> l>0
            if (tv[j] > 0.5f)  tb |= (1u << bit);
        }
    }
    masks[tid]         = pb;                             // pred maps: words [0, 32768)
    masks[32768 + tid] = tb;                             // targ maps: words [32768, 65536)

    __shared__ float red[3][256];
    red[0][threadIdx.x] = si; red[1][threadIdx.x] = sp; red[2][threadIdx.x] = st;
    __syncthreads();
    for (int s = 128; s > 0; s >>= 1) {
        if (threadIdx.x < (unsigned)s) {
            red[0][threadIdx.x] += red[0][threadIdx.x + s];
            red[1][threadIdx.x] += red[1][threadIdx.x + s];
            red[2][threadIdx.x] += red[2][threadIdx.x + s];
        }
        __syncthreads();
    }
    if (threadIdx.x == 0) {
        atomicAdd(&scal[0], red[0][0]);   // inter  = sum p*t
        atomicAdd(&scal[1], red[1][0]);   // sum p
        atomicAdd(&scal[2], red[2][0]);   // sum t
    }
}

// ---------------- kernel 2: Hough voting (WMMA fp32 16x16x4 + TDM) ------
__global__ __launch_bounds__(256) void hough_kernel(const unsigned* __restrict__ masks,
                                                    unsigned* __restrict__ ghist) {
    __shared__ unsigned hist[RHO_BINS * TCHUNK];   // 32 KB histogram
    __shared__ unsigned mrows[BAND_WORDS];         // 4 KB band bitmask
    const int blk  = blockIdx.x;
    const int band = blk % NBANDS;
    const int tc   = (blk / NBANDS) % NCHUNK;
    const int img  = blk / (NBANDS * NCHUNK);
    const int tid  = threadIdx.x;

    // Wave 0 launches the TDM DMA of this band's bitmask into LDS.
    if (tid < 32) {
        tdm_load_1d((unsigned)(uintptr_t)(void*)mrows,
                    masks + img * WORDS_PER_IMG + band * BAND_WORDS,
                    BAND_WORDS);
    }
    for (int i = tid; i < RHO_BINS * TCHUNK; i += 256) hist[i] = 0u;
    if (tid < 32) __builtin_amdgcn_s_wait_tensorcnt(0);
    __syncthreads();

    const int lane  = tid & 31;
    const int wv    = tid >> 5;
    const int tbase = tc * TCHUNK;
    const float scl = (float)(RHO_BINS - 1) / (2.0f * DIAG_F);

    // B operand (4x16), rows striped across lane halves:
    //   lanes 0-15 : vgpr0 = K0 row (cos*s), vgpr1 = K1 row (sin*s)
    //   lanes 16-31: vgpr0 = K2 row (diag*s), vgpr1 = K3 row (0)
    // theta >= NUM_THETA columns get zero rows -> they vote into dummy LDS
    // columns that are never merged (no guard needed in the hot loop).
    v2f bm;
    if (lane < 16) {
        int t = tbase + lane;
        if (t < NUM_THETA) {
            double th = -1.5707963267948966 + (double)t * (3.141592653589793 / 179.0);
            bm.x = (float)cos(th) * scl;
            bm.y = (float)sin(th) * scl;
        } else { bm.x = 0.f; bm.y = 0.f; }
    } else {
        bm.x = DIAG_F * scl;    // multiplies A's constant-1 column
        bm.y = 0.f;
    }

    const int row0 = band * ROWS_PER_BAND;
    const int n    = lane & 15;              // theta column inside chunk
    const int moff = (lane < 16) ? 0 : 8;    // C/D layout: vgpr i -> pixel M = moff+i

    // 64 rows x 32 x-tiles = 2048 tiles per block, cycled over 8 waves.
    for (int tile = wv; tile < ROWS_PER_BAND * (W_IMG / 16); tile += 8) {
        const int rloc = tile >> 5;
        const int row  = row0 + rloc;
        const int x0   = (tile & 31) << 4;
        unsigned word  = mrows[rloc * WORDS_PER_ROW + (x0 >> 5)];   // LDS, wave-uniform
        unsigned bits  = ((x0 & 16) ? (word >> 16) : word) & 0xFFFFu;
        if (__builtin_amdgcn_readfirstlane(bits) == 0u) continue;   // uniform skip

        // A operand (16x4): lanes 0-15 hold K0,K1 of pixel M=lane,
        // lanes 16-31 hold K2,K3 of pixel M=lane-16.  Rows = [x, y, 1, 0].
        v2f a;
        if (lane < 16) { a.x = (float)(x0 + lane); a.y = (float)row; }
        else           { a.x = 1.0f;               a.y = 0.0f;       }

        v8f c = {0.f, 0.f, 0.f, 0.f, 0.f, 0.f, 0.f, 0.f};
        v8f d = __builtin_amdgcn_wmma_f32_16x16x4_f32(
            /*neg_a=*/false, a, /*neg_b=*/false, bm,
            /*c_mod=*/(short)0, c, /*reuse_a=*/false, /*reuse_b=*/false);

        // Branchless vote: add the mask bit (0/1) -- no exec-mask juggling.
#pragma unroll
        for (int i = 0; i < 8; ++i) {
            unsigned vbit = (bits >> (moff + i)) & 1u;
            int idx = (int)d[i];             // trunc toward zero, like astype(int32)
            idx = idx < 0 ? 0 : (idx > RHO_BINS - 1 ? RHO_BINS - 1 : idx);
            atomicAdd(&hist[idx * TCHUNK + n], vbit);
        }
    }
    __syncthreads();

    // merge block-private histogram into global counts (skip theta >= 180)
    for (int i = tid; i < RHO_BINS * TCHUNK; i += 256) {
        unsigned v = hist[i];
        if (v) {
            int rho = i >> 4;
            int t   = tbase + (i & 15);
            if (t < NUM_THETA)
                atomicAdd(&ghist[(img * RHO_BINS + rho) * NUM_THETA + t], v);
        }
    }
}

// ---------------- kernel 3: per-map max of thresholded counts -----------
__global__ __launch_bounds__(256) void max_kernel(const unsigned* __restrict__ ghist,
                                                  unsigned* __restrict__ maxu) {
    const int img = blockIdx.x;
    const unsigned* h = ghist + img * HBINS;
    unsigned m = 0u;
    for (int i = threadIdx.x; i < HBINS; i += 256) {
        unsigned v = h[i];
        if (v >= LINE_THRESH && v > m) m = v;
    }
    __shared__ unsigned red[256];
    red[threadIdx.x] = m;
    __syncthreads();
    for (int s = 128; s > 0; s >>= 1) {
        if (threadIdx.x < (unsigned)s) {
            unsigned o = red[threadIdx.x + s];
            if (o > red[threadIdx.x]) red[threadIdx.x] = o;
        }
        __syncthreads();
    }
    if (threadIdx.x == 0) maxu[img] = red[0];
}

// ---------------- kernel 4: hough-dice partial sums ---------------------
__global__ __launch_bounds__(256) void hdice_kernel(const unsigned* __restrict__ ghist,
                                                    const unsigned* __restrict__ maxu,
                                                    float* __restrict__ scal) {
    const int gid = blockIdx.x * 256 + threadIdx.x;
    float fp = 0.f, ft = 0.f;
    if (gid < 4 * HBINS) {
        const int b = gid / HBINS, j = gid % HBINS;
        unsigned cp = ghist[b * HBINS + j];
        unsigned ct = ghist[(b + 4) * HBINS + j];
        unsigned mp = maxu[b];     if (mp == 0u) mp = 1u;   // where(m>0, m, 1)
        unsigned mt = maxu[b + 4]; if (mt == 0u) mt = 1u;
        fp = (cp >= LINE_THRESH) ? (float)cp / (float)mp : 0.f;
        ft = (ct >= LINE_THRESH) ? (float)ct / (float)mt : 0.f;
    }
    __shared__ float red[3][256];
    red[0][threadIdx.x] = fp * ft; red[1][threadIdx.x] = fp; red[2][threadIdx.x] = ft;
    __syncthreads();
    for (int s = 128; s > 0; s >>= 1) {
        if (threadIdx.x < (unsigned)s) {
            red[0][threadIdx.x] += red[0][threadIdx.x + s];
            red[1][threadIdx.x] += red[1][threadIdx.x + s];
            red[2][threadIdx.x] += red[2][threadIdx.x + s];
        }
        __syncthreads();
    }
    if (threadIdx.x == 0) {
        atomicAdd(&scal[3], red[0][0]);   // hough inter
        atomicAdd(&scal[4], red[1][0]);   // hough sum pred
        atomicAdd(&scal[5], red[2][0]);   // hough sum targ
    }
}

// ---------------- kernel 5: final scalar --------------------------------
__global__ void final_kernel(const float* __restrict__ scal, float* __restrict__ out) {
    if (threadIdx.x == 0 && blockIdx.x == 0) {
        float si = scal[0], sp = scal[1], st = scal[2];
        float score_i = 2.0f * si / fmaxf(sp + st, EPS_F);
        float loss_i  = (1.0f - score_i) * (st > 0.0f ? 1.0f : 0.0f);
        float hi = scal[3], hp = scal[4], ht = scal[5];
        float score_h = 2.0f * hi / fmaxf(hp + ht, EPS_F);
        float loss_h  = (1.0f - score_h) * (ht > 0.0f ? 1.0f : 0.0f);
        out[0] = (1.0f - ALPHA_F) * loss_i + ALPHA_F * loss_h;
    }
}

// ---------------- host launcher ----------------------------------------
extern "C" void kernel_launch(void* const* d_in, const int* in_sizes, int n_in,
                              void* d_out, int out_size, void* d_ws, size_t ws_size,
                              hipStream_t stream) {
    (void)in_sizes; (void)n_in; (void)out_size; (void)ws_size;
    const float* logits  = (const float*)d_in[0];
    const float* targets = (const float*)d_in[1];

    unsigned* ws    = (unsigned*)d_ws;
    unsigned* masks = ws;                                  // 65536 u32 (fully overwritten)
    unsigned* ghist = ws + MASK_WORDS;                     // 737280 u32
    unsigned* maxu  = ws + MASK_WORDS + GHIST_WORDS;       // 8 u32
    float*    scal  = (float*)(ws + MASK_WORDS + GHIST_WORDS + 8); // 6 f32

    const int nz = GHIST_WORDS + 8 + 6;                    // zero hist + maxes + scalars
    zero_kernel<<<(nz + 255) / 256, 256, 0, stream>>>(ghist, nz);
    prep_kernel<<<128, 256, 0, stream>>>(logits, targets, masks, scal);
    hough_kernel<<<N_MAPS * NCHUNK * NBANDS, 256, 0, stream>>>(masks, ghist);
    max_kernel<<<N_MAPS, 256, 0, stream>>>(ghist, maxu);
    hdice_kernel<<<(4 * HBINS + 255) / 256, 256, 0, stream>>>(ghist, maxu, scal);
    final_kernel<<<1, 32, 0, stream>>>(scal, (float*)d_out);
}